// LocalAttention_83717502534079
// MI455X (gfx1250) — compile-verified
//
#include <hip/hip_runtime.h>
#include <math.h>

typedef __attribute__((ext_vector_type(2))) float v2f;
typedef __attribute__((ext_vector_type(8))) float v8f;

#define S_LEN   2048
#define D_DIM   512
#define HALF    32
#define NB      5          // 5 key blocks of 16 -> 80 keys per query tile
#define LDSP    84         // LDS row stride in floats (84 mod 64 = 20, conflict-free)

__device__ __forceinline__ int clampi(int x, int lo, int hi) {
    return x < lo ? lo : (x > hi ? hi : x);
}

// ---------------------------------------------------------------------------
// Kernel 1: zero the dense (B,S,S) attention matrix (32 MiB) with float4 stores
// ---------------------------------------------------------------------------
__global__ __launch_bounds__(256) void zero_attn_kernel(float4* __restrict__ attn4, int n4) {
    int i = blockIdx.x * blockDim.x + threadIdx.x;
    if (i < n4) {
        float4 z; z.x = 0.f; z.y = 0.f; z.z = 0.f; z.w = 0.f;
        attn4[i] = z;
    }
}

// ---------------------------------------------------------------------------
// Kernel 2: one wave32 per 16-row query tile. fp32 WMMA for QK^T and PV.
// ---------------------------------------------------------------------------
__global__ __launch_bounds__(128) void local_attn_kernel(const float* __restrict__ Q,
                                                         const float* __restrict__ K,
                                                         const float* __restrict__ V,
                                                         float* __restrict__ Out,
                                                         float* __restrict__ Attn) {
    __shared__ float ldsP[4][16 * LDSP];   // per-wave 16x80 P tile (stride 84)

    const int lane = threadIdx.x & 31;
    const int warp = threadIdx.x >> 5;
    const int tile = blockIdx.x * 4 + warp;      // 256 tiles total
    const int b    = tile >> 7;                  // tile / 128
    const int q0   = (tile & 127) << 4;          // 16-row query tile base

    const int n  = lane & 15;                    // N / M position within 16
    const int hi = lane >> 4;                    // half-wave select
    const int kb = hi * 2;                       // K sub-index base for A/B frags

    const float* Qb = Q + (size_t)b * S_LEN * D_DIM;
    const float* Kb = K + (size_t)b * S_LEN * D_DIM;
    const float* Vb = V + (size_t)b * S_LEN * D_DIM;

    // key indices (per lane column n) for the 5 key blocks, plus clamped rows
    int kidx[NB], kcl[NB];
#pragma unroll
    for (int nb = 0; nb < NB; ++nb) {
        kidx[nb] = q0 - HALF + nb * 16 + n;
        kcl[nb]  = clampi(kidx[nb], 0, S_LEN - 1);
    }

    // ---------------- scores: S = Q * K^T  (5 tiles of 16x16) ----------------
    v8f acc[NB];
#pragma unroll
    for (int nb = 0; nb < NB; ++nb) acc[nb] = (v8f){};

    const float* qrow = Qb + (size_t)(q0 + n) * D_DIM + kb;   // A: row M = n
    const float* krow[NB];
#pragma unroll
    for (int nb = 0; nb < NB; ++nb) krow[nb] = Kb + (size_t)kcl[nb] * D_DIM + kb;

    for (int d4 = 0; d4 < D_DIM; d4 += 4) {
        v2f a = *(const v2f*)(qrow + d4);                     // Q[q0+n][d4+kb..+1]
#pragma unroll
        for (int nb = 0; nb < NB; ++nb) {
            v2f bf = *(const v2f*)(krow[nb] + d4);            // K[key_n][d4+kb..+1]
            acc[nb] = __builtin_amdgcn_wmma_f32_16x16x4_f32(
                false, a, false, bf, (short)0, acc[nb], false, false);
        }
    }

    // ---------------- mask + softmax over the 65-wide window ----------------
    const float scale = 0.044194173824159216f;   // 1/sqrt(512)
    const float NEG_INF = -__builtin_inff();
    float p[NB][8];

#pragma unroll
    for (int j = 0; j < 8; ++j) {
        const int m = j + 8 * hi;                // row index within tile
        float mx = NEG_INF;
#pragma unroll
        for (int nb = 0; nb < NB; ++nb) {
            int rel = -HALF + nb * 16 + n - m;   // k - s
            bool valid = (kidx[nb] >= 0) && (kidx[nb] < S_LEN) &&
                         (rel >= -HALF) && (rel <= HALF);
            float s = valid ? acc[nb][j] * scale : NEG_INF;
            p[nb][j] = s;
            mx = fmaxf(mx, s);
        }
        // row reduction across the 16 lanes holding this row
        mx = fmaxf(mx, __shfl_xor(mx, 1, 32));
        mx = fmaxf(mx, __shfl_xor(mx, 2, 32));
        mx = fmaxf(mx, __shfl_xor(mx, 4, 32));
        mx = fmaxf(mx, __shfl_xor(mx, 8, 32));

        float sum = 0.f;
#pragma unroll
        for (int nb = 0; nb < NB; ++nb) {
            float e = __expf(p[nb][j] - mx);     // exp(-inf)=0 for masked
            p[nb][j] = e;
            sum += e;
        }
        sum += __shfl_xor(sum, 1, 32);
        sum += __shfl_xor(sum, 2, 32);
        sum += __shfl_xor(sum, 4, 32);
        sum += __shfl_xor(sum, 8, 32);
        float inv = 1.0f / sum;                  // window always contains s itself
#pragma unroll
        for (int nb = 0; nb < NB; ++nb) p[nb][j] *= inv;
    }

    // ------ stage P (C-layout -> LDS row-major) + write band to Attn ------
    float* lp = ldsP[warp];
    float* attnB = Attn + (size_t)b * S_LEN * S_LEN;
#pragma unroll
    for (int j = 0; j < 8; ++j) {
        const int m = j + 8 * hi;
#pragma unroll
        for (int nb = 0; nb < NB; ++nb) {
            lp[m * LDSP + nb * 16 + n] = p[nb][j];
            if (kidx[nb] >= 0 && kidx[nb] < S_LEN)
                attnB[(size_t)(q0 + m) * S_LEN + kidx[nb]] = p[nb][j];
        }
    }
    __syncthreads();   // uniform; orders LDS writes before A-fragment reads

    // ---------------- output: O = P (16x80) * V (80x512) ----------------
    float* outRow = Out + (size_t)b * S_LEN * D_DIM;
    for (int dt = 0; dt < D_DIM / 16; ++dt) {
        const int d0 = dt * 16;
        v8f oacc = (v8f){};
        for (int kc = 0; kc < 80; kc += 4) {
            // A frag from LDS: P[row M=n][kc+kb .. +1]
            v2f a = *(const v2f*)(lp + n * LDSP + kc + kb);
            // B frag: V[key = q0-32+kc+kb+{0,1}][d0+n]  (clamped; P=0 off-band)
            int r0 = clampi(q0 - HALF + kc + kb,     0, S_LEN - 1);
            int r1 = clampi(q0 - HALF + kc + kb + 1, 0, S_LEN - 1);
            v2f bf;
            bf.x = Vb[(size_t)r0 * D_DIM + d0 + n];
            bf.y = Vb[(size_t)r1 * D_DIM + d0 + n];
            oacc = __builtin_amdgcn_wmma_f32_16x16x4_f32(
                false, a, false, bf, (short)0, oacc, false, false);
        }
#pragma unroll
        for (int j = 0; j < 8; ++j)
            outRow[(size_t)(q0 + j + 8 * hi) * D_DIM + d0 + n] = oacc[j];
    }
}

// ---------------------------------------------------------------------------
extern "C" void kernel_launch(void* const* d_in, const int* in_sizes, int n_in,
                              void* d_out, int out_size, void* d_ws, size_t ws_size,
                              hipStream_t stream) {
    (void)in_sizes; (void)n_in; (void)out_size; (void)d_ws; (void)ws_size;
    const float* Q = (const float*)d_in[0];
    const float* K = (const float*)d_in[1];
    const float* V = (const float*)d_in[2];

    const int B = 2;
    float* Out  = (float*)d_out;                              // (B,S,D)
    float* Attn = Out + (size_t)B * S_LEN * D_DIM;            // (B,S,S)

    // Zero the dense attention matrix: 2*2048*2048 floats = 2,097,152 float4s
    const int n4 = (B * S_LEN * S_LEN) / 4;
    zero_attn_kernel<<<(n4 + 255) / 256, 256, 0, stream>>>((float4*)Attn, n4);

    // 256 query tiles, 1 wave each, 4 waves per block -> 64 blocks
    local_attn_kernel<<<64, 128, 0, stream>>>(Q, K, V, Out, Attn);
}